// SamplewiseBboxTargetSelector_8358006358559
// MI455X (gfx1250) — compile-verified
//
#include <hip/hip_runtime.h>
#include <stdint.h>

#define B_ROWS   128
#define NLEN     200000
#define POS_T    0.35f
#define NEG_T    0.15f
#define K_SELECT 64u
#define POS_GOAL 32u
#define BLK      1024
#define CHUNK    1024
#define NCHUNK   ((NLEN + CHUNK - 1) / CHUNK)   // 196

// Monotone key: descending float order == descending uint order.
__device__ __forceinline__ uint32_t fkey(float f) {
  uint32_t u = __float_as_uint(f);
  return (u & 0x80000000u) ? ~u : (u | 0x80000000u);
}

// Low 32 bits of a generic pointer to __shared__ == LDS byte offset.
__device__ __forceinline__ uint32_t lds_off(const void* p) {
  return (uint32_t)(uintptr_t)p;
}

// CDNA5 async global->LDS load, saddr form (vdst = LDS byte addr, vaddr = u32
// byte offset, saddr = 64-bit base). Tracked by ASYNCcnt.
__device__ __forceinline__ void async_b32(uint32_t dst, uint64_t base, uint32_t voff) {
  asm volatile("global_load_async_to_lds_b32 %0, %1, %2"
               :: "v"(dst), "v"(voff), "s"(base) : "memory");
}

// Find bucket (scanning high->low) where cumulative count reaches k; rem = how
// many still needed inside that bucket. 1-based k, k >= 1 assumed.
__device__ __forceinline__ void pick(const uint32_t* hist, int bins, uint32_t k,
                                     uint32_t* bucket, uint32_t* rem) {
  uint32_t cum = 0;
  for (int b = bins - 1; b >= 0; --b) {
    uint32_t h = hist[b];
    if (cum + h >= k) { *bucket = (uint32_t)b; *rem = k - cum; return; }
    cum += h;
  }
  *bucket = 0; *rem = 0;
}

__global__ __launch_bounds__(BLK) void bbox_select_kernel(
    const float* __restrict__ ov, const float* __restrict__ np,
    const float* __restrict__ nn, float* __restrict__ out) {
  __shared__ uint32_t histP[2048], histN[2048];
  __shared__ float sOv[2][CHUNK], sNp[2][CHUNK], sNn[2][CHUNK];
  __shared__ uint32_t sWP[32], sWN[32];
  __shared__ uint32_t sPosCnt, sNegCnt, sKP, sKN;
  __shared__ uint32_t sPreP, sRemP, sPreN, sRemN;
  __shared__ uint32_t sKthP, sKthN, sRP, sRN, sRunP, sRunN;
  __shared__ float sBW, sSW;

  const int tid = threadIdx.x;
  const int row = blockIdx.x;
  const uint64_t ovB = (uint64_t)(uintptr_t)(ov + (size_t)row * NLEN);
  const uint64_t npB = (uint64_t)(uintptr_t)(np + (size_t)row * NLEN);
  const uint64_t nnB = (uint64_t)(uintptr_t)(nn + (size_t)row * NLEN);
  float* lab = out + (size_t)row * NLEN;
  float* scw = out + (size_t)B_ROWS * NLEN + (size_t)row * NLEN;
  float* bbw = out + 2ull * (size_t)B_ROWS * NLEN + (size_t)row * NLEN;

#define STAGE(bufi, ci) do {                                   \
    int _idx = (ci) * CHUNK + tid;                             \
    int _cl  = _idx < NLEN ? _idx : (NLEN - 1);                \
    uint32_t _vo = (uint32_t)_cl * 4u;                         \
    async_b32(lds_off(&sOv[(bufi)][tid]), ovB, _vo);           \
    async_b32(lds_off(&sNp[(bufi)][tid]), npB, _vo);           \
    async_b32(lds_off(&sNn[(bufi)][tid]), nnB, _vo);           \
  } while (0)

  // ---------------- Pass 1: counts + level-1 histograms (key bits 31:21) ----
  for (int i = tid; i < 2048; i += BLK) { histP[i] = 0; histN[i] = 0; }
  if (tid == 0) { sPosCnt = 0; sNegCnt = 0; }
  __syncthreads();

  uint32_t lp = 0, ln = 0;
  STAGE(0, 0);
  for (int c = 0; c < NCHUNK; ++c) {
    const int buf = c & 1;
    if (c + 1 < NCHUNK) { STAGE(buf ^ 1, c + 1);
                          asm volatile("s_wait_asynccnt 3" ::: "memory"); }
    else                { asm volatile("s_wait_asynccnt 0" ::: "memory"); }
    __syncthreads();
    const int idx = c * CHUNK + tid;
    if (idx < NLEN) {
      float o = sOv[buf][tid];
      if (o >= POS_T)      { ++lp; atomicAdd(&histP[fkey(sNp[buf][tid]) >> 21], 1u); }
      else if (o < NEG_T)  { ++ln; atomicAdd(&histN[fkey(sNn[buf][tid]) >> 21], 1u); }
    }
    __syncthreads();
  }
  atomicAdd(&sPosCnt, lp);
  atomicAdd(&sNegCnt, ln);
  __syncthreads();

  if (tid == 0) {
    uint32_t pc = sPosCnt, nc = sNegCnt;
    uint32_t kp = pc < POS_GOAL ? pc : POS_GOAL;
    uint32_t kn0 = K_SELECT - kp;
    uint32_t kn = nc < kn0 ? nc : kn0;
    sKP = kp; sKN = kn;
    uint32_t b, r;
    if (kp) { pick(histP, 2048, kp, &b, &r); sPreP = b; sRemP = r; }
    if (kn) { pick(histN, 2048, kn, &b, &r); sPreN = b; sRemN = r; }
    uint32_t pcc = pc ? pc : 1u;
    uint32_t tot = kp + kn; if (!tot) tot = 1u;
    sBW = 1.0f / ((float)pcc * (float)B_ROWS);
    sSW = 1.0f / ((float)tot * (float)B_ROWS);
  }
  __syncthreads();
  const uint32_t kP = sKP, kN = sKN;
  const uint32_t preP1 = kP ? sPreP : 0xFFFFFFFFu;   // 11-bit prefix sentinel-safe
  const uint32_t preN1 = kN ? sPreN : 0xFFFFFFFFu;

  // ---------------- Pass 2: refine key bits 20:10 ---------------------------
  for (int i = tid; i < 2048; i += BLK) { histP[i] = 0; histN[i] = 0; }
  __syncthreads();
  STAGE(0, 0);
  for (int c = 0; c < NCHUNK; ++c) {
    const int buf = c & 1;
    if (c + 1 < NCHUNK) { STAGE(buf ^ 1, c + 1);
                          asm volatile("s_wait_asynccnt 3" ::: "memory"); }
    else                { asm volatile("s_wait_asynccnt 0" ::: "memory"); }
    __syncthreads();
    const int idx = c * CHUNK + tid;
    if (idx < NLEN) {
      float o = sOv[buf][tid];
      if (o >= POS_T) {
        uint32_t k = fkey(sNp[buf][tid]);
        if ((k >> 21) == preP1) atomicAdd(&histP[(k >> 10) & 2047u], 1u);
      } else if (o < NEG_T) {
        uint32_t k = fkey(sNn[buf][tid]);
        if ((k >> 21) == preN1) atomicAdd(&histN[(k >> 10) & 2047u], 1u);
      }
    }
    __syncthreads();
  }
  if (tid == 0) {
    uint32_t b, r;
    if (sKP) { pick(histP, 2048, sRemP, &b, &r); sPreP = (sPreP << 11) | b; sRemP = r; }
    if (sKN) { pick(histN, 2048, sRemN, &b, &r); sPreN = (sPreN << 11) | b; sRemN = r; }
  }
  __syncthreads();
  const uint32_t preP2 = kP ? sPreP : 0xFFFFFFFFu;   // 22-bit prefix
  const uint32_t preN2 = kN ? sPreN : 0xFFFFFFFFu;

  // ---------------- Pass 3: refine key bits 9:0 -> exact kth key ------------
  for (int i = tid; i < 2048; i += BLK) { histP[i] = 0; histN[i] = 0; }
  __syncthreads();
  STAGE(0, 0);
  for (int c = 0; c < NCHUNK; ++c) {
    const int buf = c & 1;
    if (c + 1 < NCHUNK) { STAGE(buf ^ 1, c + 1);
                          asm volatile("s_wait_asynccnt 3" ::: "memory"); }
    else                { asm volatile("s_wait_asynccnt 0" ::: "memory"); }
    __syncthreads();
    const int idx = c * CHUNK + tid;
    if (idx < NLEN) {
      float o = sOv[buf][tid];
      if (o >= POS_T) {
        uint32_t k = fkey(sNp[buf][tid]);
        if ((k >> 10) == preP2) atomicAdd(&histP[k & 1023u], 1u);
      } else if (o < NEG_T) {
        uint32_t k = fkey(sNn[buf][tid]);
        if ((k >> 10) == preN2) atomicAdd(&histN[k & 1023u], 1u);
      }
    }
    __syncthreads();
  }
  if (tid == 0) {
    uint32_t b, r;
    if (sKP) { pick(histP, 1024, sRemP, &b, &r); sKthP = (sPreP << 10) | b; sRP = r; }
    else     { sKthP = 0xFFFFFFFFu; sRP = 0; }
    if (sKN) { pick(histN, 1024, sRemN, &b, &r); sKthN = (sPreN << 10) | b; sRN = r; }
    else     { sKthN = 0xFFFFFFFFu; sRN = 0; }
    sRunP = 0; sRunN = 0;
  }
  __syncthreads();
  const uint32_t kthP = sKthP, kthN = sKthN, rP = sRP, rN = sRN;
  const float bw = sBW, sw = sSW;

  // ---------------- Pass 4: stable selection + outputs ----------------------
  const int lane = tid & 31;
  const int wid  = tid >> 5;
  STAGE(0, 0);
  for (int c = 0; c < NCHUNK; ++c) {
    const int buf = c & 1;
    if (c + 1 < NCHUNK) { STAGE(buf ^ 1, c + 1);
                          asm volatile("s_wait_asynccnt 3" ::: "memory"); }
    else                { asm volatile("s_wait_asynccnt 0" ::: "memory"); }
    __syncthreads();
    const int idx = c * CHUNK + tid;
    bool isPos = false, isNeg = false;
    uint32_t kA = 0, kB = 0;
    if (idx < NLEN) {
      float o = sOv[buf][tid];
      isPos = (o >= POS_T);
      isNeg = (o < NEG_T);
      kA = fkey(sNp[buf][tid]);
      kB = fkey(sNn[buf][tid]);
    }
    const bool tieP = isPos && (kA == kthP);
    const bool hiP  = isPos && (kA >  kthP);
    const bool tieN = isNeg && (kB == kthN);
    const bool hiN  = isNeg && (kB >  kthN);
    const uint32_t mP = __builtin_amdgcn_ballot_w32(tieP);
    const uint32_t mN = __builtin_amdgcn_ballot_w32(tieN);
    if (lane == 0) { sWP[wid] = __popc(mP); sWN[wid] = __popc(mN); }
    __syncthreads();
    if (tid == 0) {  // exclusive scan over 32 wave totals, seeded by running
      uint32_t aP = sRunP, aN = sRunN;
      for (int w = 0; w < 32; ++w) {
        uint32_t tp = sWP[w]; sWP[w] = aP; aP += tp;
        uint32_t tn = sWN[w]; sWN[w] = aN; aN += tn;
      }
      sRunP = aP; sRunN = aN;
    }
    __syncthreads();
    const uint32_t lmask = (1u << lane) - 1u;
    const bool selP = hiP || (tieP && (sWP[wid] + __popc(mP & lmask) < rP));
    const bool selN = hiN || (tieN && (sWN[wid] + __popc(mN & lmask) < rN));
    if (idx < NLEN) {
      lab[idx] = isPos ? 1.0f : 0.0f;
      bbw[idx] = isPos ? bw : 0.0f;
      scw[idx] = (selP || selN) ? sw : 0.0f;
    }
    __syncthreads();
  }
#undef STAGE
}

extern "C" void kernel_launch(void* const* d_in, const int* in_sizes, int n_in,
                              void* d_out, int out_size, void* d_ws, size_t ws_size,
                              hipStream_t stream) {
  (void)in_sizes; (void)n_in; (void)out_size; (void)d_ws; (void)ws_size;
  const float* ov = (const float*)d_in[0];
  const float* np = (const float*)d_in[1];
  const float* nn = (const float*)d_in[2];
  float* out = (float*)d_out;
  bbox_select_kernel<<<dim3(B_ROWS), dim3(BLK), 0, stream>>>(ov, np, nn, out);
}